// GELU88_15255723835413
// MI455X (gfx1250) — compile-verified
//
#include <hip/hip_runtime.h>
#include <math.h>

// Vector types matching gfx1250 WMMA operand shapes
typedef __attribute__((ext_vector_type(2))) float v2f;   // f32 16x4 A / 4x16 B: 2 VGPRs
typedef __attribute__((ext_vector_type(8))) float v8f;   // f32 16x16 C/D: 8 VGPRs
typedef __attribute__((ext_vector_type(4))) float f4;    // 128-bit load/store

#define EPS_F   1e-5f
#define EPSV_F  1e-4f
#define GELU_C  0.7978845608028654f   // sqrt(2/pi)

static __device__ __forceinline__ float fast_tanh(float t) {
#if __has_builtin(__builtin_amdgcn_tanhf)
    return __builtin_amdgcn_tanhf(t);   // gfx1250 v_tanh_f32 (TRANS)
#else
    return tanhf(t);
#endif
}

// ---------------------------------------------------------------------------
// Prep: invstd[d] = 1/(sqrt(max(ema_sq - ema_mean^2, 1e-4)) + 1e-5), plus
// scalars tau / sigma / w / ez appended at ws[D..D+3].
// ---------------------------------------------------------------------------
__global__ void gelu88_prep_kernel(const float* __restrict__ log_tau,
                                   const float* __restrict__ log_sigma_raw,
                                   const float* __restrict__ log_w_raw,
                                   const float* __restrict__ ema_mean,
                                   const float* __restrict__ ema_sq,
                                   const float* __restrict__ ema_zscore,
                                   float* __restrict__ ws, int D) {
    int i = blockIdx.x * blockDim.x + threadIdx.x;
    if (i < D) {
        float mm  = ema_mean[i];
        float var = fmaxf(ema_sq[i] - mm * mm, EPSV_F);
        ws[i] = 1.0f / (sqrtf(var) + EPS_F);
    }
    if (i == 0) {
        ws[D + 0] = expf(log_tau[0]);                 // tau
        ws[D + 1] = log1pf(expf(log_sigma_raw[0]));   // sigma = softplus
        ws[D + 2] = log1pf(expf(log_w_raw[0]));       // w     = softplus
        ws[D + 3] = ema_zscore[0];                    // ez
    }
}

// ---------------------------------------------------------------------------
// Main: one 256-thread block (8 wave32) per row of D=2048.
//  - each thread: 8 contiguous cols, x streamed non-temporally
//  - per-lane f32 partials -> shfl_xor wave reduce -> 8x3 partials in LDS
//  - final cross-wave reduce + broadcast via one V_WMMA_F32_16X16X4_F32
// ---------------------------------------------------------------------------
__global__ __launch_bounds__(256) void gelu88_row_kernel(
    const float* __restrict__ x,
    const float* __restrict__ ema_mean,
    const float* __restrict__ ema_out,
    const float* __restrict__ ws,        // [0,D): invstd ; [D,D+4): scalars
    float* __restrict__ out, int D) {

    const int  tid  = threadIdx.x;
    const long base = (long)blockIdx.x * (long)D;
    const int  c0   = tid * 8;

    const float tau   = ws[D + 0];
    const float sigma = ws[D + 1];
    const float w     = ws[D + 2];
    const float ez    = ws[D + 3];

    // streamed once -> non-temporal; shared vectors -> regular (L2-resident)
    f4 xa = __builtin_nontemporal_load((const f4*)(x + base + c0));
    f4 xb = __builtin_nontemporal_load((const f4*)(x + base + c0 + 4));
    f4 ma = *(const f4*)(ema_mean + c0);
    f4 mb = *(const f4*)(ema_mean + c0 + 4);
    f4 ia = *(const f4*)(ws + c0);
    f4 ib = *(const f4*)(ws + c0 + 4);
    f4 ea = *(const f4*)(ema_out + c0);
    f4 eb = *(const f4*)(ema_out + c0 + 4);

    float xv[8], mv[8], iv[8], ev[8], od[8];
#pragma unroll
    for (int i = 0; i < 4; ++i) {
        xv[i] = xa[i]; xv[4 + i] = xb[i];
        mv[i] = ma[i]; mv[4 + i] = mb[i];
        iv[i] = ia[i]; iv[4 + i] = ib[i];
        ev[i] = ea[i]; ev[4 + i] = eb[i];
    }

    float s0 = 0.0f, s1 = 0.0f, s2 = 0.0f;   // sum|z|, sum od^2, sum od*eo
#pragma unroll
    for (int i = 0; i < 8; ++i) {
        float xx = xv[i];
        // |z| = |(x - m) * invstd|
        s0 += fabsf((xx - mv[i]) * iv[i]);
        // tanh-approx GELU: 0.5*x*(1 + tanh(C*x*(1 + 0.044715*x^2)))
        float x2 = xx * xx;
        float u  = fmaf(0.044715f, x2, 1.0f);
        float t  = GELU_C * xx * u;
        float th = fast_tanh(t);
        float hx = 0.5f * xx;
        float o  = fmaf(hx, th, hx);
        od[i] = o;
        s1 = fmaf(o, o, s1);
        s2 = fmaf(o, ev[i], s2);
    }

    // ---- wave32 reduction -------------------------------------------------
#pragma unroll
    for (int off = 16; off > 0; off >>= 1) {
        s0 += __shfl_xor(s0, off, 32);
        s1 += __shfl_xor(s1, off, 32);
        s2 += __shfl_xor(s2, off, 32);
    }

    __shared__ float part[8][3];
    __shared__ float gate_s;
    const int wave = tid >> 5;
    const int lane = tid & 31;
    if (lane == 0) { part[wave][0] = s0; part[wave][1] = s1; part[wave][2] = s2; }
    __syncthreads();

    // ---- cross-wave reduce + broadcast via one f32 WMMA -------------------
    // A (16x4 f32): lane m (0..15) holds K={0,1} in VGPRs {a0,a1};
    //               lane m+16 holds K={2,3}.  B = all-ones 4x16.
    // D[m][n] = a0[m] + a1[m] + a0[m+16] + a1[m+16]  (broadcast over n).
    // Rows 0/1/2 carry the three quantities; D VGPR q, lanes 0-15 = total S_q.
    if (tid < 32) {   // exactly wave 0: EXEC all-ones within the wave
        int  mrow = tid & 15;
        int  side = tid >> 4;
        float a0 = 0.0f, a1 = 0.0f;
        if (mrow < 3) {
            int b = side * 4;
            a0 = part[b + 0][mrow] + part[b + 1][mrow];
            a1 = part[b + 2][mrow] + part[b + 3][mrow];
        }
        v2f A; A[0] = a0;   A[1] = a1;
        v2f Bm; Bm[0] = 1.0f; Bm[1] = 1.0f;
        v8f C = {};
        C = __builtin_amdgcn_wmma_f32_16x16x4_f32(
                /*neg_a=*/false, A, /*neg_b=*/false, Bm,
                /*c_mod=*/(short)0, C, /*reuse_a=*/false, /*reuse_b=*/false);
        if (tid == 0) {
            float S0 = C[0], S1 = C[1], S2 = C[2];
            float abs_mean = S0 * (1.0f / (float)D);
            float rel      = abs_mean / (ez + EPS_F);
            float surp     = fast_tanh(sigma * (rel - 1.0f));
            float nrm      = fmaxf(sqrtf(S1), 1e-12f);
            float cosv     = S2 / nrm;
            gate_s = expf(-tau * cosv) * fmaf(w, surp, 1.0f);
        }
    }
    __syncthreads();

    const float g = gate_s;
    f4 oa, ob;
#pragma unroll
    for (int i = 0; i < 4; ++i) { oa[i] = od[i] * g; ob[i] = od[4 + i] * g; }
    __builtin_nontemporal_store(oa, (f4*)(out + base + c0));
    __builtin_nontemporal_store(ob, (f4*)(out + base + c0 + 4));
}

// ---------------------------------------------------------------------------
extern "C" void kernel_launch(void* const* d_in, const int* in_sizes, int n_in,
                              void* d_out, int out_size, void* d_ws, size_t ws_size,
                              hipStream_t stream) {
    const float* x             = (const float*)d_in[0];
    const float* log_tau       = (const float*)d_in[1];
    const float* log_sigma_raw = (const float*)d_in[2];
    const float* log_w_raw     = (const float*)d_in[3];
    const float* ema_mean      = (const float*)d_in[4];
    const float* ema_sq        = (const float*)d_in[5];
    const float* ema_out       = (const float*)d_in[6];
    const float* ema_zscore    = (const float*)d_in[7];
    float*       out           = (float*)d_out;
    float*       ws            = (float*)d_ws;   // D invstd floats + 4 scalars

    const int D    = in_sizes[4];          // 2048
    const int rows = in_sizes[0] / D;      // B*T = 16384

    gelu88_prep_kernel<<<(D + 255) / 256, 256, 0, stream>>>(
        log_tau, log_sigma_raw, log_w_raw, ema_mean, ema_sq, ema_zscore, ws, D);

    gelu88_row_kernel<<<rows, 256, 0, stream>>>(
        x, ema_mean, ema_out, ws, out, D);
}